// SSIM_52793738002531
// MI455X (gfx1250) — compile-verified
//
#include <hip/hip_runtime.h>
#include <hip/hip_bf16.h>

// ---------------------------------------------------------------------------
// SSIM on MI455X (gfx1250).  Bandwidth-bound (~100MB inputs @ 23.3 TB/s).
// Fused tile kernel: LDS-staged patches, separable Gaussian conv with the
// vertical pass done as  D = W(16x28) x H(28x16) + C  on V_WMMA_F32_16X16X4_F32
// (7 K-steps of 4), per the CDNA5 ISA 32-bit A/B/C/D wave32 layouts.
// ---------------------------------------------------------------------------

typedef __attribute__((ext_vector_type(2))) float v2f;
typedef __attribute__((ext_vector_type(8))) float v8f;

#define IMG_H 512
#define IMG_W 512
#define OUT_HW 502
#define N_IMG 48              // 16 batch * 3 channels
#define TILES_PER_IMG 1024    // 32 x 32 tiles of 16x16 outputs
#define WAVES_PER_BLOCK 8
#define PATCH_W 26
#define PATCH_ROWS 28         // 26 live rows + 2 zero rows (K pad 26 -> 28)
#define PATCH_ELEMS (PATCH_ROWS * PATCH_W)

// Gaussian weight value g[d] for d in [0,10]; 0.26601172 = 1/sum = g[5].
__device__ __forceinline__ float gw(int d) {
  if (d < 0 || d > 10) return 0.0f;
  float x = (float)(d - 5);
  return 0.26601172f * __expf(x * x * (-1.0f / 4.5f));
}

__global__ void init_ws(unsigned* wsu) {
  wsu[0] = 0x7F800000u;          // min accumulator: +inf bits
  wsu[1] = 0u;                   // max accumulator: 0.0f (inputs >= 0)
  ((float*)wsu)[2] = 0.0f;       // ssim sum
}

__global__ __launch_bounds__(256) void minmax_k(const float4* __restrict__ x,
                                                int n4, unsigned* wsu) {
  int tid = blockIdx.x * blockDim.x + threadIdx.x;
  int stride = gridDim.x * blockDim.x;
  float4 v = x[tid];
  float mn = fminf(fminf(v.x, v.y), fminf(v.z, v.w));
  float mx = fmaxf(fmaxf(v.x, v.y), fmaxf(v.z, v.w));
  for (int i = tid + stride; i < n4; i += stride) {
    v = x[i];
    mn = fminf(mn, fminf(fminf(v.x, v.y), fminf(v.z, v.w)));
    mx = fmaxf(mx, fmaxf(fmaxf(v.x, v.y), fmaxf(v.z, v.w)));
  }
  #pragma unroll
  for (int off = 16; off > 0; off >>= 1) {
    mn = fminf(mn, __shfl_xor(mn, off, 32));
    mx = fmaxf(mx, __shfl_xor(mx, off, 32));
  }
  if ((threadIdx.x & 31) == 0) {
    // values are non-negative -> float bit pattern is monotone as uint
    atomicMin(&wsu[0], __float_as_uint(mn));
    atomicMax(&wsu[1], __float_as_uint(mx));
  }
}

__global__ __launch_bounds__(256) void ssim_k(const float* __restrict__ P,
                                              const float* __restrict__ T,
                                              unsigned* __restrict__ wsu) {
  // Gaussian taps (sigma=1.5, 11 wide, normalized); unrolled -> literals.
  constexpr float G[11] = {
      0.00102838f, 0.00759876f, 0.03600077f, 0.10936069f, 0.21300553f,
      0.26601172f, 0.21300553f, 0.10936069f, 0.03600077f, 0.00759876f,
      0.00102838f};

  __shared__ float shP[WAVES_PER_BLOCK][PATCH_ELEMS];
  __shared__ float shT[WAVES_PER_BLOCK][PATCH_ELEMS];

  const int lane = threadIdx.x & 31;
  const int w    = threadIdx.x >> 5;
  const int n    = lane & 15;     // B/D column, and A row (M)
  const int half = lane >> 4;     // lane half selects K sub-pair / M+8

  const int tileIdx = blockIdx.x * WAVES_PER_BLOCK + w;
  const int img = tileIdx >> 10;            // / TILES_PER_IMG
  const int rem = tileIdx & 1023;
  const int y0 = (rem >> 5) << 4;           // tile origin in 502x502 output
  const int x0 = (rem & 31) << 4;

  const float* Pi = P + (size_t)img * (IMG_H * IMG_W);
  const float* Ti = T + (size_t)img * (IMG_H * IMG_W);

  // ---- cooperative patch load: 26x26 live, rows 26..27 forced zero ----
  for (int f = lane; f < PATCH_ELEMS; f += 32) {
    int r = f / PATCH_W;
    int c = f - r * PATCH_W;
    int gy = y0 + r, gx = x0 + c;
    float pv = 0.0f, tv = 0.0f;
    if (r < 26 && gy < IMG_H && gx < IMG_W) {
      int idx = gy * IMG_W + gx;
      pv = Pi[idx];
      tv = Ti[idx];
    }
    shP[w][f] = pv;
    shT[w][f] = tv;
  }
  __syncthreads();

  // ---- A operand: banded Gaussian matrix W[m][j] = g[j-m], 16x28.
  // f32 16x4 A layout: lanes 0-15 hold K={4i,4i+1}, lanes 16-31 K={4i+2,4i+3}.
  v2f aW[7];
  #pragma unroll
  for (int i = 0; i < 7; i++) {
    int j0 = 4 * i + 2 * half;
    aW[i].x = gw(j0 - n);
    aW[i].y = gw(j0 + 1 - n);
  }

  v8f accP  = {0, 0, 0, 0, 0, 0, 0, 0};   // mu1
  v8f accT  = {0, 0, 0, 0, 0, 0, 0, 0};   // mu2
  v8f accPP = {0, 0, 0, 0, 0, 0, 0, 0};   // E[p^2]
  v8f accTT = {0, 0, 0, 0, 0, 0, 0, 0};   // E[t^2]
  v8f accPT = {0, 0, 0, 0, 0, 0, 0, 0};   // E[p t]

  // ---- 7 K-groups: horizontal 11-tap pass for rows 4i+2*half+{0,1},
  // producing B operands directly in the f32 4x16 B layout, then 5 WMMAs.
  #pragma unroll
  for (int i = 0; i < 7; i++) {
    const int r0 = 4 * i + 2 * half;
    const float* rP0 = &shP[w][r0 * PATCH_W + n];
    const float* rT0 = &shT[w][r0 * PATCH_W + n];
    float hp0 = 0, hp1 = 0, ht0 = 0, ht1 = 0;
    float hpp0 = 0, hpp1 = 0, htt0 = 0, htt1 = 0, hpt0 = 0, hpt1 = 0;
    #pragma unroll
    for (int j = 0; j < 11; j++) {
      const float gj = G[j];
      float p0 = rP0[j];
      float t0 = rT0[j];
      float p1 = rP0[j + PATCH_W];
      float t1 = rT0[j + PATCH_W];
      hp0  += gj * p0;        ht0  += gj * t0;
      hpp0 += gj * (p0 * p0); htt0 += gj * (t0 * t0); hpt0 += gj * (p0 * t0);
      hp1  += gj * p1;        ht1  += gj * t1;
      hpp1 += gj * (p1 * p1); htt1 += gj * (t1 * t1); hpt1 += gj * (p1 * t1);
    }
    v2f b;
    b.x = hp0;  b.y = hp1;
    accP  = __builtin_amdgcn_wmma_f32_16x16x4_f32(false, aW[i], false, b,
                                                  (short)0, accP,  false, false);
    b.x = ht0;  b.y = ht1;
    accT  = __builtin_amdgcn_wmma_f32_16x16x4_f32(false, aW[i], false, b,
                                                  (short)0, accT,  false, false);
    b.x = hpp0; b.y = hpp1;
    accPP = __builtin_amdgcn_wmma_f32_16x16x4_f32(false, aW[i], false, b,
                                                  (short)0, accPP, false, false);
    b.x = htt0; b.y = htt1;
    accTT = __builtin_amdgcn_wmma_f32_16x16x4_f32(false, aW[i], false, b,
                                                  (short)0, accTT, false, false);
    b.x = hpt0; b.y = hpt1;
    accPT = __builtin_amdgcn_wmma_f32_16x16x4_f32(false, aW[i], false, b,
                                                  (short)0, accPT, false, false);
  }

  // ---- SSIM map + masked partial sum.  C/D layout: VGPR e -> M = e + 8*half.
  const float L  = __uint_as_float(wsu[1]) - __uint_as_float(wsu[0]);
  const float C1 = (0.01f * L) * (0.01f * L);
  const float C2 = (0.03f * L) * (0.03f * L);

  float sum = 0.0f;
  #pragma unroll
  for (int e = 0; e < 8; e++) {
    int oy = y0 + e + 8 * half;
    int ox = x0 + n;
    if (oy < OUT_HW && ox < OUT_HW) {
      float mu1 = accP[e], mu2 = accT[e];
      float mu1s = mu1 * mu1, mu2s = mu2 * mu2, m12 = mu1 * mu2;
      float s1  = accPP[e] - mu1s;
      float s2  = accTT[e] - mu2s;
      float s12 = accPT[e] - m12;
      float v1 = 2.0f * s12 + C2;
      float v2 = s1 + s2 + C2;
      sum += ((2.0f * m12 + C1) * v1) / ((mu1s + mu2s + C1) * v2);
    }
  }
  #pragma unroll
  for (int off = 16; off > 0; off >>= 1) sum += __shfl_xor(sum, off, 32);
  if (lane == 0) atomicAdd((float*)(wsu + 2), sum);
}

__global__ void finalize_k(const unsigned* wsu, float* out) {
  const float total = (float)N_IMG * (float)OUT_HW * (float)OUT_HW;  // 12096192
  out[0] = -(((const float*)wsu)[2] / total);
}

extern "C" void kernel_launch(void* const* d_in, const int* in_sizes, int n_in,
                              void* d_out, int out_size, void* d_ws, size_t ws_size,
                              hipStream_t stream) {
  const float* y_pred = (const float*)d_in[0];
  const float* y_true = (const float*)d_in[1];
  float* out = (float*)d_out;
  unsigned* wsu = (unsigned*)d_ws;

  const int n = in_sizes[0];                 // 16*3*512*512 = 12,582,912

  init_ws<<<1, 1, 0, stream>>>(wsu);
  minmax_k<<<1024, 256, 0, stream>>>((const float4*)y_pred, n / 4, wsu);

  const int n_tiles  = N_IMG * TILES_PER_IMG;            // 49152
  const int n_blocks = n_tiles / WAVES_PER_BLOCK;        // 6144
  ssim_k<<<n_blocks, 256, 0, stream>>>(y_pred, y_true, wsu);

  finalize_k<<<1, 1, 0, stream>>>(wsu, out);
}